// CubeSimulator_69355131896681
// MI455X (gfx1250) — compile-verified
//
#include <hip/hip_runtime.h>

typedef __attribute__((ext_vector_type(2))) float v2f;
typedef __attribute__((ext_vector_type(8))) float v8f;

namespace {
constexpr int   IRES   = 128;                     // internal spatial resolution
constexpr int   VRES   = 80;                      // internal velocity channels
constexpr float FOV    = 1000.0f;                 // pc
constexpr float M2PC   = 1.0f / 3.086e16f;
constexpr float VMAXPC = 200000.0f * M2PC;        // rotation-curve amplitude (pc/s)
constexpr float RC     = 0.1f  * FOV;             // core radius
constexpr float RD     = 0.3f  * FOV;             // disk scale length
constexpr float HZ     = 0.05f * FOV;             // vertical scale height
constexpr float VLO    = -300000.0f * M2PC;       // channel range (pc/s)
constexpr float VHI    =  300000.0f * M2PC;
constexpr float LOG2E  = 1.4426950408889634f;
}

// One block per output spatial cell (io,jo): 4x4 input pixels, full K=128 depth,
// all 80 channels. 256 threads = 8 wave32.
__global__ __launch_bounds__(256) void cube_kde_wmma(
    const float* __restrict__ p_incl,
    const float* __restrict__ p_srot,
    const float* __restrict__ p_lb,
    float* __restrict__ out)
{
    // {vz, src} interleaved so phase B reads 2 k-steps per ds_load_b128.
    __shared__ __align__(16) float2 s_field[16][IRES];
    __shared__ float s_cube[VRES][16];  // cube[v, pixel] for this tile

    const int tid = threadIdx.x;
    const int io  = blockIdx.x >> 5;    // output row    (0..31)
    const int jo  = blockIdx.x & 31;    // output column (0..31)

    // Uniform scalar parameters
    const float incl = p_incl[0];
    const float srot = p_srot[0];
    const float lb   = p_lb[0];
    const float ci = __cosf(incl), si = __sinf(incl);
    const float cr = __cosf(srot), sr = __sinf(srot);
    const float sig2 = lb * lb;
    const float A2   = -LOG2E / sig2;   // exp(-d^2/s^2) == exp2(A2*d^2), A2 < 0
    const float norm = rsqrtf(2.0f * 3.14159265358979f * sig2) * (1.0f / 16.0f);

    const float step = (2.0f * FOV) / (float)(IRES - 1);

    // ---------- Phase A: field generation (trig-free inner math) ----------
    // vz = -sin(i)*v_abs*cos(theta), cos(theta) = rot_x/r2d  (no atan2/sin/cos)
    // src = exp(-(r2d/RD + rz^2/(2 HZ^2)))                   (single exp)
    #pragma unroll
    for (int m = 0; m < 8; ++m) {
        const int idx = tid + 256 * m;       // 2048 points = 16 px * 128 k
        const int p   = idx >> 7;
        const int k   = idx & (IRES - 1);
        const int gi  = io * 4 + (p >> 2);
        const int gj  = jo * 4 + (p & 3);
        const float x = -FOV + step * (float)gi;
        const float y = -FOV + step * (float)gj;
        const float z = -FOV + step * (float)k;
        const float x1 = cr * x - sr * y;
        const float y1 = sr * x + cr * y;
        const float rx = x1;
        const float ry = ci * y1 - si * z;
        const float rz = si * y1 + ci * z;
        const float r2d2 = rx * rx + ry * ry;
        const float r3d2 = r2d2 + rz * rz;
        const float r3d  = __fsqrt_rn(r3d2);
        const float vabs = VMAXPC * r3d * __frsqrt_rn(r3d2 + RC * RC);
        const float r2d  = __fsqrt_rn(r2d2);
        const float ir2d = (r2d > 0.0f) ? (1.0f / r2d) : 0.0f;
        const float vz   = -si * vabs * rx * ir2d;
        const float srcv = __expf(-(r2d * (1.0f / RD) + rz * rz * (0.5f / (HZ * HZ))));
        s_field[p][k] = make_float2(vz, srcv);
    }
    __syncthreads();

    // ---------- Phase B: KDE reduction (the 1.68e8-exp core) ----------
    // thread -> (pixel p, channels c0, c0+16, ..., c0+64).
    // A2*(zl-vz)^2 expanded as fma(m_c, vz, fma(A2*vz, vz, czl_c)):
    // per channel: 2 fma + v_exp_f32 + acc-fma -> trans(exp)-bound loop.
    const int   p     = tid & 15;
    const int   c0    = tid >> 4;
    const float zstep = (VHI - VLO) / (float)(VRES - 1);
    const float zl0 = VLO + zstep * (float)(c0);
    const float zl1 = VLO + zstep * (float)(c0 + 16);
    const float zl2 = VLO + zstep * (float)(c0 + 32);
    const float zl3 = VLO + zstep * (float)(c0 + 48);
    const float zl4 = VLO + zstep * (float)(c0 + 64);
    const float cz0 = A2 * zl0 * zl0, m0 = -2.0f * A2 * zl0;
    const float cz1 = A2 * zl1 * zl1, m1 = -2.0f * A2 * zl1;
    const float cz2 = A2 * zl2 * zl2, m2 = -2.0f * A2 * zl2;
    const float cz3 = A2 * zl3 * zl3, m3 = -2.0f * A2 * zl3;
    const float cz4 = A2 * zl4 * zl4, m4 = -2.0f * A2 * zl4;
    float a0 = 0.f, a1 = 0.f, a2 = 0.f, a3 = 0.f, a4 = 0.f;

    const float4* row = reinterpret_cast<const float4*>(&s_field[p][0]);
    #pragma unroll 4
    for (int k2 = 0; k2 < IRES / 2; ++k2) {
        const float4 f = row[k2];            // {vz0, src0, vz1, src1}
        {
            const float vz = f.x, s = f.y;
            const float q  = A2 * vz;
            a0 = __builtin_fmaf(__builtin_amdgcn_exp2f(__builtin_fmaf(m0, vz, __builtin_fmaf(q, vz, cz0))), s, a0);
            a1 = __builtin_fmaf(__builtin_amdgcn_exp2f(__builtin_fmaf(m1, vz, __builtin_fmaf(q, vz, cz1))), s, a1);
            a2 = __builtin_fmaf(__builtin_amdgcn_exp2f(__builtin_fmaf(m2, vz, __builtin_fmaf(q, vz, cz2))), s, a2);
            a3 = __builtin_fmaf(__builtin_amdgcn_exp2f(__builtin_fmaf(m3, vz, __builtin_fmaf(q, vz, cz3))), s, a3);
            a4 = __builtin_fmaf(__builtin_amdgcn_exp2f(__builtin_fmaf(m4, vz, __builtin_fmaf(q, vz, cz4))), s, a4);
        }
        {
            const float vz = f.z, s = f.w;
            const float q  = A2 * vz;
            a0 = __builtin_fmaf(__builtin_amdgcn_exp2f(__builtin_fmaf(m0, vz, __builtin_fmaf(q, vz, cz0))), s, a0);
            a1 = __builtin_fmaf(__builtin_amdgcn_exp2f(__builtin_fmaf(m1, vz, __builtin_fmaf(q, vz, cz1))), s, a1);
            a2 = __builtin_fmaf(__builtin_amdgcn_exp2f(__builtin_fmaf(m2, vz, __builtin_fmaf(q, vz, cz2))), s, a2);
            a3 = __builtin_fmaf(__builtin_amdgcn_exp2f(__builtin_fmaf(m3, vz, __builtin_fmaf(q, vz, cz3))), s, a3);
            a4 = __builtin_fmaf(__builtin_amdgcn_exp2f(__builtin_fmaf(m4, vz, __builtin_fmaf(q, vz, cz4))), s, a4);
        }
    }
    s_cube[c0     ][p] = a0;
    s_cube[c0 + 16][p] = a1;
    s_cube[c0 + 32][p] = a2;
    s_cube[c0 + 48][p] = a3;
    s_cube[c0 + 64][p] = a4;
    __syncthreads();

    // ---------- Phase C: channel pooling as a real GEMM on the matrix core ----------
    // D[16 vo x 16 px] = P[16 x 80] * cube[80 x 16], P[vo,v] = 0.2 iff v/5 == vo.
    // Exact in f32 via chained V_WMMA_F32_16X16X4_F32 (20 steps of K=4), wave 0 only
    // (full wave enters the branch -> EXEC all ones as WMMA requires).
    if (tid < 32) {
        const int lane = tid;
        const int half = lane >> 4;      // K-slot split {0,1} vs {2,3}
        const int m16  = lane & 15;      // A row (vo) and B column (pixel)
        const int lo   = 5 * m16;        // nonzero band of P for this row
        const int hi   = lo + 5;
        v8f acc = {};
        for (int vb = 0; vb < VRES; vb += 4) {
            const int v0 = vb + 2 * half;
            const int v1 = v0 + 1;
            v2f A, B;
            A.x = (v0 >= lo && v0 < hi) ? 0.2f : 0.0f;
            A.y = (v1 >= lo && v1 < hi) ? 0.2f : 0.0f;
            B.x = s_cube[v0][m16];
            B.y = s_cube[v1][m16];
            acc = __builtin_amdgcn_wmma_f32_16x16x4_f32(
                false, A, false, B, (short)0, acc, false, false);
        }
        // Spatial mean: reduce D over N (16 pixels) within each half-wave,
        // then lane 0 / lane 16 store rows M=r / M=r+8. Exclusive writes -> no atomics.
        #pragma unroll
        for (int r = 0; r < 8; ++r) {
            float v = acc[r];
            v += __shfl_xor(v, 1);
            v += __shfl_xor(v, 2);
            v += __shfl_xor(v, 4);
            v += __shfl_xor(v, 8);
            if ((lane & 15) == 0) {
                const int vo = r + 8 * half;
                out[(vo << 10) + (io << 5) + jo] = v * norm;
            }
        }
    }
}

extern "C" void kernel_launch(void* const* d_in, const int* in_sizes, int n_in,
                              void* d_out, int out_size, void* d_ws, size_t ws_size,
                              hipStream_t stream) {
    (void)in_sizes; (void)n_in; (void)out_size; (void)d_ws; (void)ws_size;
    const float* incl = (const float*)d_in[0];
    const float* srot = (const float*)d_in[1];
    const float* lb   = (const float*)d_in[2];
    float* out = (float*)d_out;
    // 32x32 output cells, one block each; every out element written exactly once.
    hipLaunchKernelGGL(cube_kde_wmma, dim3(32 * 32), dim3(256), 0, stream,
                       incl, srot, lb, out);
}